// BiGRUModel_17660905521351
// MI455X (gfx1250) — compile-verified
//
#include <hip/hip_runtime.h>

typedef __attribute__((ext_vector_type(2))) float v2f;
typedef __attribute__((ext_vector_type(4))) float v4f;
typedef __attribute__((ext_vector_type(8))) float v8f;
typedef __attribute__((ext_vector_type(4))) unsigned int v4u;
typedef __attribute__((ext_vector_type(4))) int v4i;
typedef __attribute__((ext_vector_type(8))) int v8i;

#define H_DIM 64
#define D_DIM 16
#define NG 192
#define CHUNK 32
#define CHUNK_F (CHUNK * D_DIM)          // 512 floats per row per chunk
#define XBUF_F (16 * CHUNK_F)            // 8192 floats per buffer

#if defined(__has_builtin)
#if __has_builtin(__builtin_amdgcn_tensor_load_to_lds) && \
    __has_builtin(__builtin_amdgcn_s_wait_tensorcnt)
#define USE_TDM 1
#endif
#endif
#ifndef USE_TDM
#define USE_TDM 0
#endif

__device__ __forceinline__ v8f wmma4(v2f a, v2f b, v8f c) {
  // D = A(16x4 f32) * B(4x16 f32) + C(16x16 f32)
  return __builtin_amdgcn_wmma_f32_16x16x4_f32(false, a, false, b, (short)0, c,
                                               false, false);
}

__device__ __forceinline__ v8f v8zero() {
  v8f z;
#pragma unroll
  for (int i = 0; i < 8; ++i) z[i] = 0.0f;
  return z;  // constant-folds -> WMMA SRC2 inline 0
}

__device__ __forceinline__ float fsigmoid(float x) {
  return 1.0f / (1.0f + __expf(-x));
}
__device__ __forceinline__ float ftanh_fast(float x) {
  // tanh(x) = 1 - 2/(exp(2x)+1); safe at +-inf
  return 1.0f - 2.0f / (1.0f + __expf(2.0f * x));
}

#if USE_TDM
// TDM: DMA a 2-D tile (16 rows x 512 f32, row stride = stride_elems f32)
// from global memory into LDS at lds_byte_addr. ISA D# per cdna5 §8.3/8.4.
__device__ __forceinline__ void tdm_load_chunk(const float* gsrc,
                                               unsigned lds_byte_addr,
                                               unsigned stride_elems) {
  unsigned long long ga = (unsigned long long)(uintptr_t)gsrc;
  v4u g0;
  g0[0] = 1u;                                   // count=1, user descriptor
  g0[1] = lds_byte_addr;                        // lds_addr
  g0[2] = (unsigned)ga;                         // global_addr[31:0]
  g0[3] = (unsigned)((ga >> 32) & 0x01FFFFFFu)  // global_addr[56:32]
          | (2u << 30);                         // type = 2 ("image")
  const unsigned TD0 = CHUNK_F;                 // tensor_dim0 (f32 units)
  const unsigned TD1 = 16u;                     // tensor_dim1 (rows)
  const unsigned TL0 = CHUNK_F;                 // tile_dim0
  const unsigned TL1 = 16u;                     // tile_dim1
  v8i g1;
  g1[0] = (int)(2u << 16);                      // wg_mask=0, data_size=4B
  g1[1] = (int)((TD0 & 0xFFFFu) << 16);         // abar=0 | tensor_dim0 lo16
  g1[2] = (int)((TD0 >> 16) | ((TD1 & 0xFFFFu) << 16));
  g1[3] = (int)((TD1 >> 16) | (TL0 << 16));     // tensor_dim1 hi | tile_dim0
  g1[4] = (int)(TL1 & 0xFFFFu);                 // tile_dim1 | tile_dim2=0
  g1[5] = (int)stride_elems;                    // tensor_dim0_stride lo32
  g1[6] = 0;                                    // stride hi16 | dim1_stride lo
  g1[7] = 0;
  v4i z4 = {0, 0, 0, 0};
#if __clang_major__ >= 23
  v8i z8;
#pragma unroll
  for (int i = 0; i < 8; ++i) z8[i] = 0;
  __builtin_amdgcn_tensor_load_to_lds(g0, g1, z4, z4, z8, 0);
#else
  __builtin_amdgcn_tensor_load_to_lds(g0, g1, z4, z4, 0);
#endif
}
#endif

__global__ __launch_bounds__(128) void bigru_kernel(
    const float* __restrict__ x,
    const float* __restrict__ w_ih_f, const float* __restrict__ w_hh_f,
    const float* __restrict__ b_ih_f, const float* __restrict__ b_hh_f,
    const float* __restrict__ w_ih_b,
    const float* __restrict__ b_ih_b, const float* __restrict__ b_hh_b,
    const float* __restrict__ fc_w, const float* __restrict__ fc_b,
    float* __restrict__ out, int T) {
  // LDS: h[1024] | gates 4x[1024] | w_ih frags 3x[1024] | xbuf 2x[8192]
  __shared__ float smem[1024 + 4096 + 3072 + 2 * XBUF_F];
  float* Hlds = smem;
  float* Gl = smem + 1024;
  float* wfrag = smem + 1024 + 4096;
  float* xbuf = smem + 1024 + 4096 + 3072;   // double-buffered TDM target

  const int tid = threadIdx.x;
  const int wid = tid >> 5;  // wave: 0=r, 1=z, 2=hn, 3=xn
  const int lane = tid & 31;
  const int l16 = lane & 15;
  const int lhi = lane >> 4;
  const int gb = blockIdx.x;  // 16-row batch tile
  const long xrowstride = (long)T * D_DIM;
  const float* xblk = x + (long)gb * 16 * xrowstride;

  // ---- resident w_hh fragments (loaded once, reused T times) ----
  v2f Bh[16][4];  // B[k][n] = w_hh[gate*64+n][k]
  float biasv[4];
  {
    const int gateh = (wid < 3) ? wid : 0;
#pragma unroll
    for (int kk = 0; kk < 16; ++kk)
#pragma unroll
      for (int tile = 0; tile < 4; ++tile) {
        int row = gateh * H_DIM + tile * 16 + l16;
        Bh[kk][tile] = *reinterpret_cast<const v2f*>(w_hh_f + row * H_DIM +
                                                     kk * 4 + 2 * lhi);
      }
#pragma unroll
    for (int tile = 0; tile < 4; ++tile) {
      int col = tile * 16 + l16;
      float bv;
      if (wid == 0)      bv = b_ih_f[col] + b_hh_f[col];
      else if (wid == 1) bv = b_ih_f[H_DIM + col] + b_hh_f[H_DIM + col];
      else if (wid == 2) bv = b_hh_f[2 * H_DIM + col];
      else               bv = b_ih_f[2 * H_DIM + col];
      biasv[tile] = bv;
    }
  }

  // ---- w_ih fragments -> LDS table: wfrag[gate][kk][tile][lane*2+j] ----
#pragma unroll
  for (int i = 0; i < 24; ++i) {
    int idx = tid + 128 * i;  // 3072 = 3 gates * 1024
    int lanej = idx & 63;
    int tile = (idx >> 6) & 3;
    int kk = (idx >> 8) & 3;
    int g = idx >> 10;
    int l = lanej >> 1, j = lanej & 1;
    wfrag[idx] = w_ih_f[(g * H_DIM + tile * 16 + (l & 15)) * D_DIM + 4 * kk +
                        2 * (l >> 4) + j];
  }

  // h0 = 0
#pragma unroll
  for (int i = 0; i < 8; ++i) Hlds[tid * 8 + i] = 0.0f;

#if USE_TDM
  // kick off DMA of chunk 0 into buffer 0 (wave 0 only; EXEC-independent)
  if (wid == 0)
    tdm_load_chunk(xblk, (unsigned)(uintptr_t)xbuf, (unsigned)xrowstride);
#endif
  __syncthreads();

  // elementwise-phase index precompute (fragment order: tile*256 + lane*8 + reg)
  const int etile = tid >> 5;
  const int eln = tid & 31;
  const int ebase = etile * 256 + eln * 8;
  const int ejcol = etile * 16 + (eln & 15);
  const int ebrow0 = 8 * (eln >> 4);
  // LDS base for this wave's w_ih fragments (wave 3 owns gate n = index 2)
  const int gatex = (wid == 3) ? 2 : wid;
  const float* wfx = wfrag + gatex * 1024 + lane * 2;

  for (int t = 0; t < T; ++t) {
    const int tt = t & (CHUNK - 1);
    const float* xb = xbuf + ((t >> 5) & 1) * XBUF_F;  // current buffer
    if (tt == 0) {
#if USE_TDM
      if (wid == 0) __builtin_amdgcn_s_wait_tensorcnt(0);  // chunk arrived
      __syncthreads();
      if (wid == 0 && t + CHUNK < T) {  // prefetch next chunk -> other buffer
        unsigned sel = ((t >> 5) + 1) & 1;
        tdm_load_chunk(xblk + (long)(t + CHUNK) * D_DIM,
                       (unsigned)(uintptr_t)xbuf + sel * (XBUF_F * 4u),
                       (unsigned)xrowstride);
      }
#else
      // fallback: cooperative staging of x[:, t..t+CHUNK, :] into xb
#pragma unroll 4
      for (int i = 0; i < 16; ++i) {
        int q = tid + 128 * i;
        int row = q >> 7;
        int rem = q & 127;
        v4f v = *reinterpret_cast<const v4f*>(xblk + (long)row * xrowstride +
                                              (long)t * D_DIM + rem * 4);
        *reinterpret_cast<v4f*>(const_cast<float*>(xb) + row * CHUNK_F +
                                rem * 4) = v;
      }
      __syncthreads();
#endif
    }

    // ---- per-gate GEMM, two partial accumulators (8 independent WMMA
    //      chains); both chains start from WMMA's inline-0 C operand.
    v8f acc[4], acc2[4];

    if (wid < 3) {
      {  // kk pair 0/1 starts both chains from inline 0
        v2f a0 = *reinterpret_cast<const v2f*>(Hlds + l16 * H_DIM + 2 * lhi);
        v2f a1 =
            *reinterpret_cast<const v2f*>(Hlds + l16 * H_DIM + 4 + 2 * lhi);
#pragma unroll
        for (int tile = 0; tile < 4; ++tile)
          acc[tile] = wmma4(a0, Bh[0][tile], v8zero());
#pragma unroll
        for (int tile = 0; tile < 4; ++tile)
          acc2[tile] = wmma4(a1, Bh[1][tile], v8zero());
      }
#pragma unroll
      for (int kk = 1; kk < 8; ++kk) {
        v2f a0 = *reinterpret_cast<const v2f*>(Hlds + l16 * H_DIM +
                                               (2 * kk) * 4 + 2 * lhi);
        v2f a1 = *reinterpret_cast<const v2f*>(Hlds + l16 * H_DIM +
                                               (2 * kk + 1) * 4 + 2 * lhi);
#pragma unroll
        for (int tile = 0; tile < 4; ++tile)
          acc[tile] = wmma4(a0, Bh[2 * kk][tile], acc[tile]);
#pragma unroll
        for (int tile = 0; tile < 4; ++tile)
          acc2[tile] = wmma4(a1, Bh[2 * kk + 1][tile], acc2[tile]);
      }
    }

    if (wid != 2) {  // input projection, K=16: kk 0,1 -> acc ; kk 2,3 -> acc2
#pragma unroll
      for (int kk = 0; kk < 4; ++kk) {
        v2f a = *reinterpret_cast<const v2f*>(xb + l16 * CHUNK_F + tt * D_DIM +
                                              kk * 4 + 2 * lhi);
#pragma unroll
        for (int tile = 0; tile < 4; ++tile) {
          v2f bx =
              *reinterpret_cast<const v2f*>(wfx + (kk * 4 + tile) * H_DIM);
          if (wid == 3) {  // x-only wave: chains start from inline 0
            if (kk == 0)      acc[tile]  = wmma4(a, bx, v8zero());
            else if (kk == 1) acc[tile]  = wmma4(a, bx, acc[tile]);
            else if (kk == 2) acc2[tile] = wmma4(a, bx, v8zero());
            else              acc2[tile] = wmma4(a, bx, acc2[tile]);
          } else {
            if (kk < 2)
              acc[tile] = wmma4(a, bx, acc[tile]);
            else
              acc2[tile] = wmma4(a, bx, acc2[tile]);
          }
        }
      }
    }

    // ---- publish gates to LDS (combine partials + bias; r,z sigmoided) ----
    {
      float* garr = Gl + wid * 1024;
      const bool dosig = (wid < 2);
#pragma unroll
      for (int tile = 0; tile < 4; ++tile) {
        v4f lo, hi;
#pragma unroll
        for (int i = 0; i < 4; ++i) {
          float v0 = acc[tile][i] + acc2[tile][i] + biasv[tile];
          float v1 = acc[tile][i + 4] + acc2[tile][i + 4] + biasv[tile];
          lo[i] = dosig ? fsigmoid(v0) : v0;
          hi[i] = dosig ? fsigmoid(v1) : v1;
        }
        *reinterpret_cast<v4f*>(garr + tile * 256 + lane * 8) = lo;
        *reinterpret_cast<v4f*>(garr + tile * 256 + lane * 8 + 4) = hi;
      }
    }
    __syncthreads();

    // ---- h' = (1-z)*tanh(xn + r*hn) + z*h, split across all 4 waves ----
    {
      const float* Rg = Gl;
      const float* Zg = Gl + 1024;
      const float* HNg = Gl + 2048;
      const float* XNg = Gl + 3072;
#pragma unroll
      for (int i = 0; i < 8; ++i) {
        int e = ebase + i;
        float r = Rg[e], z = Zg[e], hn = HNg[e], xn = XNg[e];
        float n = ftanh_fast(xn + r * hn);
        int hidx = (ebrow0 + i) * H_DIM + ejcol;
        float hold = Hlds[hidx];
        Hlds[hidx] = (1.0f - z) * n + z * hold;
      }
    }
    __syncthreads();
  }

  // ---- backward direction collapses to ONE step from h0=0 (only ys_b[0]
  //      is used), so w_hh_b contributes only its bias. Then the FC. ----
  {
    const float* xlast =
        xbuf + (((T >> 5) - 1) & 1) * XBUF_F;  // buffer of last chunk
#pragma unroll
    for (int i = 0; i < 24; ++i) {
      int idx = tid + 128 * i;  // 3072 = 16 rows * 192 gates
      int row = idx / NG;
      int g = idx - row * NG;
      const float* wv = w_ih_b + g * D_DIM;
      const float* xv = xlast + row * CHUNK_F + (CHUNK - 1) * D_DIM;
      float s = b_ih_b[g];
#pragma unroll
      for (int d = 0; d < D_DIM; ++d) s += xv[d] * wv[d];
      Gl[idx] = s;
    }
  }
  __syncthreads();
  if (tid < 16) {
    int row = tid;
    float acc = fc_b[0];
    for (int j = 0; j < H_DIM; ++j) {
      float r = fsigmoid(Gl[row * NG + j] + b_hh_b[j]);
      float z = fsigmoid(Gl[row * NG + H_DIM + j] + b_hh_b[H_DIM + j]);
      float n = ftanh_fast(Gl[row * NG + 2 * H_DIM + j] +
                           r * b_hh_b[2 * H_DIM + j]);
      float hb = (1.0f - z) * n;  // z*h0 = 0
      acc += fc_w[j] * Hlds[row * H_DIM + j] + fc_w[H_DIM + j] * hb;
    }
    out[gb * 16 + row] = acc;
  }
}

extern "C" void kernel_launch(void* const* d_in, const int* in_sizes, int n_in,
                              void* d_out, int out_size, void* d_ws,
                              size_t ws_size, hipStream_t stream) {
  const float* x      = (const float*)d_in[0];
  const float* w_ih_f = (const float*)d_in[1];
  const float* w_hh_f = (const float*)d_in[2];
  const float* b_ih_f = (const float*)d_in[3];
  const float* b_hh_f = (const float*)d_in[4];
  const float* w_ih_b = (const float*)d_in[5];
  // d_in[6] = w_hh_b: multiplies h0==0 in the single backward step used -> unused
  const float* b_ih_b = (const float*)d_in[7];
  const float* b_hh_b = (const float*)d_in[8];
  const float* fc_w   = (const float*)d_in[9];
  const float* fc_b   = (const float*)d_in[10];
  float* out = (float*)d_out;

  const int T = 512;
  const int B = in_sizes[0] / (T * D_DIM);  // 512
  bigru_kernel<<<dim3(B / 16), dim3(128), 0, stream>>>(
      x, w_ih_f, w_hh_f, b_ih_f, b_hh_f, w_ih_b, b_ih_b, b_hh_b, fc_w, fc_b,
      out, T);
}